// EncoderTreeNN_21844203668321
// MI455X (gfx1250) — compile-verified
//
#include <hip/hip_runtime.h>
#include <hip/hip_bf16.h>

#define HOPS  3
#define VOCAB 50257
#define DDIM  128
#define NB    32      // batch
#define NM    512     // story slots
#define NS    6       // tokens per slot
#define NT    128     // trees
#define NL    64      // tokens per tree
#define MT    (NM + NT)   // 640 memories
#define OUT_CHUNKS 8

typedef __attribute__((ext_vector_type(2))) float v2f;
typedef __attribute__((ext_vector_type(8))) float v8f;

// ---------------------------------------------------------------- zero u
__global__ void zero_u_kernel(float* __restrict__ u) {
  int i = blockIdx.x * blockDim.x + threadIdx.x;
  if (i < NB * DDIM) u[i] = 0.0f;
}

// ---------------------------------------------------------------- scores
// grid = (MT/16, B), block = 512 (16 waves). Wave w builds the embedding-sum
// row for memory slot m0+w into LDS; wave 0 then runs the 16x16x4 f32 WMMA
// chain against u[b] (replicated across N) and extracts scores from D col 0.
__global__ __launch_bounds__(512)
void score_kernel(const float* __restrict__ Ct, const int* __restrict__ story,
                  const int* __restrict__ kb, const float* __restrict__ u,
                  float* __restrict__ scores) {
  __shared__ float tile[16][DDIM];           // 8 KB
  const int b    = blockIdx.y;
  const int m0   = blockIdx.x * 16;
  const int w    = threadIdx.x >> 5;
  const int lane = threadIdx.x & 31;
  const int m    = m0 + w;
  const int dbase = lane * 4;

  float4 acc = make_float4(0.f, 0.f, 0.f, 0.f);
  if (m < NM) {
    const int* toks = story + ((size_t)b * NM + m) * NS;
    #pragma unroll
    for (int s = 0; s < NS; ++s) {
      const float4 r = *(const float4*)(Ct + ((size_t)toks[s] << 7) + dbase);
      acc.x += r.x; acc.y += r.y; acc.z += r.z; acc.w += r.w;
    }
  } else {
    const int* toks = kb + ((size_t)b * NT + (m - NM)) * NL;
    for (int s = 0; s < NL; ++s) {
      const float4 r = *(const float4*)(Ct + ((size_t)toks[s] << 7) + dbase);
      acc.x += r.x; acc.y += r.y; acc.z += r.z; acc.w += r.w;
    }
  }
  *(float4*)&tile[w][dbase] = acc;
  __syncthreads();

  if (threadIdx.x < 32) {                    // wave 0, EXEC all-ones
    const float* ub  = u + (size_t)b * DDIM;
    const int mrow = lane & 15;
    const int khi  = (lane >> 4) ? 2 : 0;    // lanes 16-31 hold K=2,3
    v8f c = {};
    for (int k = 0; k < DDIM; k += 4) {
      v2f a, bv;
      a[0]  = tile[mrow][k + khi];
      a[1]  = tile[mrow][k + khi + 1];
      bv[0] = ub[k + khi];                   // all 16 N-columns identical
      bv[1] = ub[k + khi + 1];
      c = __builtin_amdgcn_wmma_f32_16x16x4_f32(
              false, a, false, bv, (short)0, c, false, false);
    }
    // D 16x16 f32: lane n<16 holds M=0..7 (vgpr 0..7) of column n,
    // lane 16+n holds M=8..15. All columns equal; read column 0.
    if ((lane & 15) == 0) {
      const int mbase = m0 + ((lane >> 4) ? 8 : 0);
      float* sb = scores + (size_t)b * MT + mbase;
      #pragma unroll
      for (int r = 0; r < 8; ++r) sb[r] = c[r];
    }
  }
}

// ---------------------------------------------------------------- softmax
// grid = B blocks, 256 threads; softmax over the 640 scores in place.
__global__ __launch_bounds__(256)
void softmax_kernel(float* __restrict__ scores) {
  __shared__ float red[256];
  float* s = scores + (size_t)blockIdx.x * MT;
  const int tid = threadIdx.x;

  float mx = -3.0e38f;
  for (int i = tid; i < MT; i += 256) mx = fmaxf(mx, s[i]);
  red[tid] = mx; __syncthreads();
  for (int off = 128; off > 0; off >>= 1) {
    if (tid < off) red[tid] = fmaxf(red[tid], red[tid + off]);
    __syncthreads();
  }
  mx = red[0]; __syncthreads();

  float sum = 0.f;
  for (int i = tid; i < MT; i += 256) {
    float e = __expf(s[i] - mx);
    s[i] = e;
    sum += e;
  }
  red[tid] = sum; __syncthreads();
  for (int off = 128; off > 0; off >>= 1) {
    if (tid < off) red[tid] += red[tid + off];
    __syncthreads();
  }
  const float inv = 1.0f / red[0];
  for (int i = tid; i < MT; i += 256) s[i] *= inv;
}

// ---------------------------------------------------------------- weighted sum (partials)
// grid = (OUT_CHUNKS, B), block 512. Each block handles MT/OUT_CHUNKS slots,
// gathers C[hop+1] rows, accumulates prob-weighted sums, LDS-reduces the 16
// wave partials, writes a [chunk][b][D] partial (no atomics -> deterministic).
__global__ __launch_bounds__(512)
void out_partial_kernel(const float* __restrict__ Ct, const int* __restrict__ story,
                        const int* __restrict__ kb, const float* __restrict__ prob,
                        float* __restrict__ part) {
  __shared__ float tile[16][DDIM];
  const int chunk = blockIdx.x;
  const int b     = blockIdx.y;
  const int w     = threadIdx.x >> 5;
  const int lane  = threadIdx.x & 31;
  const int dbase = lane * 4;
  const int mlo   = chunk * (MT / OUT_CHUNKS);
  const int mhi   = mlo + (MT / OUT_CHUNKS);
  const float* pb = prob + (size_t)b * MT;

  float4 acc = make_float4(0.f, 0.f, 0.f, 0.f);
  for (int m = mlo + w; m < mhi; m += 16) {
    const float p = pb[m];
    if (m < NM) {
      const int* toks = story + ((size_t)b * NM + m) * NS;
      #pragma unroll
      for (int s = 0; s < NS; ++s) {
        const float4 r = *(const float4*)(Ct + ((size_t)toks[s] << 7) + dbase);
        acc.x += p * r.x; acc.y += p * r.y; acc.z += p * r.z; acc.w += p * r.w;
      }
    } else {
      const int* toks = kb + ((size_t)b * NT + (m - NM)) * NL;
      for (int s = 0; s < NL; ++s) {
        const float4 r = *(const float4*)(Ct + ((size_t)toks[s] << 7) + dbase);
        acc.x += p * r.x; acc.y += p * r.y; acc.z += p * r.z; acc.w += p * r.w;
      }
    }
  }
  *(float4*)&tile[w][dbase] = acc;
  __syncthreads();
  if (threadIdx.x < DDIM) {
    float sum = 0.f;
    #pragma unroll
    for (int ww = 0; ww < 16; ++ww) sum += tile[ww][threadIdx.x];
    part[((size_t)chunk * NB + b) * DDIM + threadIdx.x] = sum;
  }
}

// ---------------------------------------------------------------- reduce partials into u
__global__ __launch_bounds__(128)
void out_reduce_kernel(const float* __restrict__ part, float* __restrict__ u) {
  const int b = blockIdx.x;
  const int d = threadIdx.x;
  float sum = 0.f;
  #pragma unroll
  for (int c = 0; c < OUT_CHUNKS; ++c)
    sum += part[((size_t)c * NB + b) * DDIM + d];
  u[(size_t)b * DDIM + d] += sum;
}

// ---------------------------------------------------------------- launcher
extern "C" void kernel_launch(void* const* d_in, const int* in_sizes, int n_in,
                              void* d_out, int out_size, void* d_ws, size_t ws_size,
                              hipStream_t stream) {
  (void)in_sizes; (void)n_in; (void)out_size; (void)ws_size;
  const float* C     = (const float*)d_in[0];   // [4][VOCAB][128]
  const int*   story = (const int*)d_in[1];     // [32][512][6]
  const int*   kb    = (const int*)d_in[2];     // [32][128][64]
  float* u = (float*)d_out;                     // [32][128] = final output

  float* scores = (float*)d_ws;                                 // 32*640 f32 (80 KB)
  float* part   = scores + (size_t)NB * MT;                     // 8*32*128 f32 (128 KB)

  zero_u_kernel<<<dim3((NB * DDIM + 255) / 256), 256, 0, stream>>>(u);

  for (int hop = 0; hop < HOPS; ++hop) {
    const float* CA = C + (size_t)hop       * VOCAB * DDIM;
    const float* CC = C + (size_t)(hop + 1) * VOCAB * DDIM;
    score_kernel<<<dim3(MT / 16, NB), 512, 0, stream>>>(CA, story, kb, u, scores);
    softmax_kernel<<<dim3(NB), 256, 0, stream>>>(scores);
    out_partial_kernel<<<dim3(OUT_CHUNKS, NB), 512, 0, stream>>>(CC, story, kb, scores, part);
    out_reduce_kernel<<<dim3(NB), 128, 0, stream>>>(part, u);
  }
}